// Net_11519102287942
// MI455X (gfx1250) — compile-verified
//
#include <hip/hip_runtime.h>
#include <hip/hip_bf16.h>
#include <math.h>

typedef __attribute__((ext_vector_type(2))) float v2f;
typedef __attribute__((ext_vector_type(8))) float v8f;

// ---------------------------------------------------------------------------
// Kernel 1: zero an int/float region of workspace (determinism across replays)
// ---------------------------------------------------------------------------
__global__ void zero_i32_kernel(int* __restrict__ p, long n) {
  long i = (long)blockIdx.x * blockDim.x + threadIdx.x;
  if (i < n) p[i] = 0;
}

// ---------------------------------------------------------------------------
// Kernel 2: degree counts via hardware int atomics (L2-resident counters)
// ---------------------------------------------------------------------------
__global__ void degree_kernel(const int* __restrict__ src, const int* __restrict__ dst,
                              int* __restrict__ deg_out, int* __restrict__ deg_in, int E) {
  int e = blockIdx.x * blockDim.x + threadIdx.x;
  if (e < E) {
    atomicAdd(&deg_out[src[e]], 1);
    atomicAdd(&deg_in[dst[e]], 1);
  }
}

// ---------------------------------------------------------------------------
// Kernel 3: hscaled = relu(x) * rsqrt(max(deg_out,1)); float4 per thread
// ---------------------------------------------------------------------------
__global__ void scale_kernel(const float* __restrict__ x, const int* __restrict__ deg_out,
                             float* __restrict__ hscaled, int N) {
  int t = blockIdx.x * blockDim.x + threadIdx.x;
  int n = t >> 2;
  int c = (t & 3) << 2;
  if (n < N) {
    float rs = rsqrtf(fmaxf((float)deg_out[n], 1.0f));
    float4 v = *(const float4*)(x + (size_t)n * 16 + c);
    v.x = fmaxf(v.x, 0.0f) * rs;
    v.y = fmaxf(v.y, 0.0f) * rs;
    v.z = fmaxf(v.z, 0.0f) * rs;
    v.w = fmaxf(v.w, 0.0f) * rs;
    *(float4*)(hscaled + (size_t)n * 16 + c) = v;
  }
}

// ---------------------------------------------------------------------------
// Kernel 4: edge scatter-add SpMM. 4 lanes per edge, float4 gather, hardware
// global_atomic_add_f32 scatter. hscaled/agg (6.4MB each) stay in 192MB L2.
// ---------------------------------------------------------------------------
__global__ void scatter_kernel(const int* __restrict__ src, const int* __restrict__ dst,
                               const float* __restrict__ hscaled, float* __restrict__ agg,
                               int E) {
  long t = (long)blockIdx.x * blockDim.x + threadIdx.x;
  int e = (int)(t >> 2);
  int c = ((int)t & 3) << 2;
  if (e < E) {
    int s = src[e];
    int d = dst[e];
    float4 v = *(const float4*)(hscaled + (size_t)s * 16 + c);
    float* ap = agg + (size_t)d * 16 + c;
    unsafeAtomicAdd(ap + 0, v.x);   // global_atomic_add_f32 (no CAS loop)
    unsafeAtomicAdd(ap + 1, v.y);
    unsafeAtomicAdd(ap + 2, v.z);
    unsafeAtomicAdd(ap + 3, v.w);
  }
}

// ---------------------------------------------------------------------------
// Kernel 5: out1 = (agg * rsqrt(deg_in)) @ W2 + b2 via V_WMMA_F32_16X16X4_F32
// One wave per 16-node tile; K=16 accumulated as four 16x16x4 WMMAs.
// A layout (16x4 f32): lanes 0-15 hold row (lane&15) K={4c,4c+1},
//                      lanes 16-31 hold row (lane&15) K={4c+2,4c+3}.
// D layout: VGPR v, lanes 0-15 -> (row v, col lane); lanes 16-31 -> (row v+8).
// ---------------------------------------------------------------------------
__global__ __launch_bounds__(256)
void conv_wmma_kernel(const float* __restrict__ agg, const int* __restrict__ deg_in,
                      const float* __restrict__ W2, const float* __restrict__ b2,
                      float* __restrict__ out, int N, int n_tiles) {
  int wave = (blockIdx.x * blockDim.x + threadIdx.x) >> 5;
  int lane = threadIdx.x & 31;
  if (wave >= n_tiles) return;              // wave-uniform guard: EXEC all-1s at WMMA
  int half = lane >> 4;
  int l16  = lane & 15;
  int row  = wave * 16 + l16;
  int rc   = row < N ? row : N - 1;
  float rs = rsqrtf(fmaxf((float)deg_in[rc], 1.0f));

  v2f a[4], b[4];
#pragma unroll
  for (int ch = 0; ch < 4; ++ch) {
    int k = ch * 4 + half * 2;
    const float* ar = agg + (size_t)rc * 16 + k;
    a[ch].x = ar[0] * rs;
    a[ch].y = ar[1] * rs;
    b[ch].x = W2[(k + 0) * 16 + l16];
    b[ch].y = W2[(k + 1) * 16 + l16];
  }
  v8f acc = {};
#pragma unroll
  for (int ch = 0; ch < 4; ++ch)
    acc = __builtin_amdgcn_wmma_f32_16x16x4_f32(false, a[ch], false, b[ch],
                                                (short)0, acc, false, false);
  float bias = b2[l16];
#pragma unroll
  for (int v = 0; v < 8; ++v) {
    int m = v + half * 8;
    int orow = wave * 16 + m;
    if (orow < N) out[(size_t)orow * 16 + l16] = acc[v] + bias;
  }
}

// ---------------------------------------------------------------------------
// Kernel 6: MLP head. h2 = concat(x, ext) [2N,16].
//   t = relu(h2 @ Wd1(pad 16x16) + bd1)   -> WMMA f32 16x16x4 (K=16)
//   z = t @ Wd2[8x2] + bd2; logits = log_softmax(z)
// Second layer done per-lane from an LDS transpose of the WMMA D tile.
// ---------------------------------------------------------------------------
__global__ __launch_bounds__(256)
void mlp_wmma_kernel(const float* __restrict__ x, const float* __restrict__ ext,
                     const float* __restrict__ Wd1, const float* __restrict__ bd1,
                     const float* __restrict__ Wd2, const float* __restrict__ bd2,
                     float* __restrict__ out, int N, int n_rows, int n_tiles) {
  __shared__ float lds[8][16 * 17];         // 8 waves/block, 16x16 tile, padded
  int wib  = threadIdx.x >> 5;
  int wave = blockIdx.x * (blockDim.x >> 5) + wib;
  int lane = threadIdx.x & 31;
  if (wave >= n_tiles) return;              // wave-uniform guard
  int half = lane >> 4;
  int l16  = lane & 15;
  int row  = wave * 16 + l16;
  int rc   = row < n_rows ? row : n_rows - 1;
  const float* hrow = (rc < N) ? (x + (size_t)rc * 16) : (ext + (size_t)(rc - N) * 16);

  v2f a[4], b[4];
#pragma unroll
  for (int ch = 0; ch < 4; ++ch) {
    int k = ch * 4 + half * 2;
    a[ch].x = hrow[k];
    a[ch].y = hrow[k + 1];
    // Wd1 is [16,8]; pad output columns 8..15 with zeros
    b[ch].x = (l16 < 8) ? Wd1[(k + 0) * 8 + l16] : 0.0f;
    b[ch].y = (l16 < 8) ? Wd1[(k + 1) * 8 + l16] : 0.0f;
  }
  v8f acc = {};
#pragma unroll
  for (int ch = 0; ch < 4; ++ch)
    acc = __builtin_amdgcn_wmma_f32_16x16x4_f32(false, a[ch], false, b[ch],
                                                (short)0, acc, false, false);
  float bias = (l16 < 8) ? bd1[l16] : 0.0f;
#pragma unroll
  for (int v = 0; v < 8; ++v) {
    int m = v + half * 8;
    lds[wib][m * 17 + l16] = fmaxf(acc[v] + bias, 0.0f);   // relu
  }
  // Wave-local ordering of DS store -> DS load (cross-lane transpose)
  __builtin_amdgcn_fence(__ATOMIC_RELEASE, "wavefront");
  __builtin_amdgcn_wave_barrier();
  __builtin_amdgcn_fence(__ATOMIC_ACQUIRE, "wavefront");

  if (half == 0) {                          // lanes 0-15: one output row each
    int r = wave * 16 + l16;
    if (r < n_rows) {
      float z0 = bd2[0], z1 = bd2[1];
#pragma unroll
      for (int k = 0; k < 8; ++k) {
        float tv = lds[wib][l16 * 17 + k];
        z0 = fmaf(tv, Wd2[k * 2 + 0], z0);
        z1 = fmaf(tv, Wd2[k * 2 + 1], z1);
      }
      float mx  = fmaxf(z0, z1);
      float lse = mx + logf(expf(z0 - mx) + expf(z1 - mx));
      out[(size_t)r * 2 + 0] = z0 - lse;
      out[(size_t)r * 2 + 1] = z1 - lse;
    }
  }
}

// ---------------------------------------------------------------------------
extern "C" void kernel_launch(void* const* d_in, const int* in_sizes, int n_in,
                              void* d_out, int out_size, void* d_ws, size_t ws_size,
                              hipStream_t stream) {
  const float* x   = (const float*)d_in[0];
  const float* ext = (const float*)d_in[1];
  const int*   src = (const int*)d_in[2];
  const int*   dst = (const int*)d_in[3];
  const float* W2  = (const float*)d_in[4];
  const float* b2  = (const float*)d_in[5];
  const float* Wd1 = (const float*)d_in[6];
  const float* bd1 = (const float*)d_in[7];
  const float* Wd2 = (const float*)d_in[8];
  const float* bd2 = (const float*)d_in[9];

  const int N = in_sizes[0] / 16;
  const int E = in_sizes[2];
  const int n_rows = 2 * N;

  // Workspace layout
  int*   deg_out = (int*)d_ws;                       // N ints
  int*   deg_in  = deg_out + N;                      // N ints
  float* hscaled = (float*)(deg_in + N);             // N*16 f32
  float* agg     = hscaled + (size_t)N * 16;         // N*16 f32

  float* out_conv   = (float*)d_out;                 // N*16
  float* out_logits = out_conv + (size_t)N * 16;     // 2N*2

  const int B = 256;

  // 1) zero degree counters + agg accumulator (fresh every call)
  long ndeg = 2L * N;
  zero_i32_kernel<<<(int)((ndeg + B - 1) / B), B, 0, stream>>>(deg_out, ndeg);
  long nagg = 16L * N;
  zero_i32_kernel<<<(int)((nagg + B - 1) / B), B, 0, stream>>>((int*)agg, nagg);

  // 2) degrees
  degree_kernel<<<(E + B - 1) / B, B, 0, stream>>>(src, dst, deg_out, deg_in, E);

  // 3) hscaled = relu(x) * rsqrt(deg_out)
  int nscale = N * 4;
  scale_kernel<<<(nscale + B - 1) / B, B, 0, stream>>>(x, deg_out, hscaled, N);

  // 4) agg[dst] += hscaled[src]   (4 lanes/edge, f32 atomics)
  long nscat = (long)E * 4;
  scatter_kernel<<<(int)((nscat + B - 1) / B), B, 0, stream>>>(src, dst, hscaled, agg, E);

  // 5) conv output via WMMA: one wave per 16-node tile
  int conv_tiles = (N + 15) / 16;
  int conv_blocks = (conv_tiles * 32 + B - 1) / B;
  conv_wmma_kernel<<<conv_blocks, B, 0, stream>>>(agg, deg_in, W2, b2, out_conv, N, conv_tiles);

  // 6) MLP + log_softmax via WMMA: one wave per 16-row tile over 2N rows
  int mlp_tiles = (n_rows + 15) / 16;
  int mlp_blocks = (mlp_tiles * 32 + B - 1) / B;
  mlp_wmma_kernel<<<mlp_blocks, B, 0, stream>>>(x, ext, Wd1, bd1, Wd2, bd2,
                                                out_logits, N, n_rows, mlp_tiles);
}